// LConvBilin_60009283060064
// MI455X (gfx1250) — compile-verified
//
#include <hip/hip_runtime.h>

typedef float v2f __attribute__((ext_vector_type(2)));
typedef float v8f __attribute__((ext_vector_type(8)));

#define NS_TOTAL 16384          // B * X  (2 * 8192)
#define SITES    8              // sites per workgroup (8192 % 8 == 0, never crosses b)
#define NCOL     (SITES * 18)   // 144 GEMM columns  (site-major, 9 complex * 2 parts)
#define KPAD     112            // 109 t_f channels padded to multiple of 4
#define MPAD     80             // 78 (= 6*13) weight rows padded to multiple of 16
#define THREADS  256

struct C3 { float2 m[9]; };     // row-major complex 3x3, m[i*3+j]

__device__ inline float2 cmul(float2 a, float2 b) {
  return make_float2(a.x*b.x - a.y*b.y, a.x*b.y + a.y*b.x);
}
__device__ inline float2 cmulcj(float2 a, float2 b) {   // a * conj(b)
  return make_float2(a.x*b.x + a.y*b.y, a.y*b.x - a.x*b.y);
}
__device__ inline float2 cjmul(float2 a, float2 b) {    // conj(a) * b
  return make_float2(a.x*b.x + a.y*b.y, a.x*b.y - a.y*b.x);
}

// lattice DIMS = (8,8,8,16), strides (1024,128,16,1); dir = +1 / -1
__device__ inline int neighbor(int s, int axis, int dir) {
  int a0 = (s >> 10) & 7, a1 = (s >> 7) & 7, a2 = (s >> 4) & 7, a3 = s & 15;
  if      (axis == 0) a0 = (a0 + dir) & 7;
  else if (axis == 1) a1 = (a1 + dir) & 7;
  else if (axis == 2) a2 = (a2 + dir) & 7;
  else                a3 = (a3 + dir) & 15;
  return (a0 << 10) | (a1 << 7) | (a2 << 4) | a3;
}

// x layout: (bx, ch[10], i, j, part) -> ((bx*10 + ch)*9 + i*3+j)*2 + p
__device__ inline void loadMat(const float* __restrict__ xin, int bx, int ch, C3& M) {
  const float* p = xin + ((size_t)bx * 10 + ch) * 18;
  #pragma unroll
  for (int e = 0; e < 9; ++e) M.m[e] = make_float2(p[2*e], p[2*e + 1]);
}

__device__ inline void mul33(const C3& A, const C3& B, C3& C) {
  #pragma unroll
  for (int i = 0; i < 3; ++i)
    #pragma unroll
    for (int k = 0; k < 3; ++k) {
      float2 acc = make_float2(0.f, 0.f);
      #pragma unroll
      for (int j = 0; j < 3; ++j) {
        float2 t = cmul(A.m[i*3+j], B.m[j*3+k]);
        acc.x += t.x; acc.y += t.y;
      }
      C.m[i*3+k] = acc;
    }
}

// R = U * W * U^dagger   (forward transport)
__device__ inline void sandwichF(const C3& U, const C3& W, C3& R) {
  C3 T; mul33(U, W, T);
  #pragma unroll
  for (int i = 0; i < 3; ++i)
    #pragma unroll
    for (int l = 0; l < 3; ++l) {
      float2 acc = make_float2(0.f, 0.f);
      #pragma unroll
      for (int k = 0; k < 3; ++k) {
        float2 t = cmulcj(T.m[i*3+k], U.m[l*3+k]);
        acc.x += t.x; acc.y += t.y;
      }
      R.m[i*3+l] = acc;
    }
}

// R = U^dagger * W * U   (backward transport)
__device__ inline void sandwichB(const C3& U, const C3& W, C3& R) {
  C3 T;
  #pragma unroll
  for (int i = 0; i < 3; ++i)
    #pragma unroll
    for (int k = 0; k < 3; ++k) {
      float2 acc = make_float2(0.f, 0.f);
      #pragma unroll
      for (int j = 0; j < 3; ++j) {
        float2 t = cjmul(U.m[j*3+i], W.m[j*3+k]);
        acc.x += t.x; acc.y += t.y;
      }
      T.m[i*3+k] = acc;
    }
  #pragma unroll
  for (int i = 0; i < 3; ++i)
    #pragma unroll
    for (int l = 0; l < 3; ++l) {
      float2 acc = make_float2(0.f, 0.f);
      #pragma unroll
      for (int k = 0; k < 3; ++k) {
        float2 t = cmul(T.m[i*3+k], U.m[k*3+l]);
        acc.x += t.x; acc.y += t.y;
      }
      R.m[i*3+l] = acc;
    }
}

__device__ inline void storeRow(float* dst, int ch, int ls, const C3& M) {
  float* p = dst + ch * NCOL + ls * 18;
  #pragma unroll
  for (int e = 0; e < 9; ++e) { p[2*e] = M.m[e].x; p[2*e + 1] = M.m[e].y; }
}
__device__ inline void storeRowDag(float* dst, int ch, int ls, const C3& M) {
  float* p = dst + ch * NCOL + ls * 18;
  #pragma unroll
  for (int i = 0; i < 3; ++i)
    #pragma unroll
    for (int j = 0; j < 3; ++j) {
      p[(i*3+j)*2]     =  M.m[j*3+i].x;
      p[(i*3+j)*2 + 1] = -M.m[j*3+i].y;
    }
}

__global__ __launch_bounds__(THREADS)
void lconv_bilin_kernel(const float* __restrict__ xin,
                        const float* __restrict__ wgt,
                        float* __restrict__ out) {
  extern __shared__ float smem[];
  float* Wt = smem;                   // MPAD*KPAD  = 8960 floats
  float* tF = Wt + MPAD * KPAD;       // KPAD*NCOL  = 16128 floats
  float* rL = tF + KPAD * NCOL;       // MPAD*NCOL  = 11520 floats
  float* wF = rL + MPAD * NCOL;       // 13*NCOL    = 1872 floats

  const int tid  = threadIdx.x;
  const int base = blockIdx.x * SITES;   // first global bx of this block

  // ---- stage padded weight matrix (78x109 -> 80x112) ----
  for (int idx = tid; idx < MPAD * KPAD; idx += THREADS) {
    int m_ = idx / KPAD, k_ = idx % KPAD;
    Wt[idx] = (m_ < 78 && k_ < 109) ? wgt[m_ * 109 + k_] : 0.0f;
  }
  // zero K-padding rows of tF (channels 109..111)
  for (int idx = tid; idx < 3 * NCOL; idx += THREADS) tF[109 * NCOL + idx] = 0.0f;

  // ---- Phase A: transports -> t_f channels 6..53 (+ daggers 60..107) ----
  for (int t = tid; t < SITES * 48; t += THREADS) {
    int ls = t / 48, rem = t % 48;
    int axis = rem / 12, rr = rem % 12;
    int c = rr % 6;
    bool backward = (rr < 6);          // term order: o = -1 first, then +1
    int bx = base + ls;
    int s = bx & 8191, bofs = bx & ~8191;
    C3 U, W, R;
    if (backward) {                    // U(x-a)^† W(x-a) U(x-a)
      int sn = neighbor(s, axis, -1);
      loadMat(xin, bofs | sn, axis, U);
      loadMat(xin, bofs | sn, 4 + c, W);
      sandwichB(U, W, R);
    } else {                           // U(x) W(x+a) U(x)^†
      int sn = neighbor(s, axis, +1);
      loadMat(xin, bx, axis, U);
      loadMat(xin, bofs | sn, 4 + c, W);
      sandwichF(U, W, R);
    }
    int ch = 6 + axis * 12 + rr;
    storeRow(tF, ch, ls, R);
    storeRowDag(tF, 54 + ch, ls, R);
  }

  // local w -> t_f[0..5], dagger -> t_f[54..59]; also w_f[0..5], w_f[6..11]
  for (int t = tid; t < SITES * 6; t += THREADS) {
    int ls = t / 6, c = t % 6;
    C3 W; loadMat(xin, base + ls, 4 + c, W);
    storeRow(tF, c, ls, W);
    storeRowDag(tF, 54 + c, ls, W);
    storeRow(wF, c, ls, W);
    storeRowDag(wF, 6 + c, ls, W);
  }

  // identity channels: t_f[108], w_f[12]
  for (int t = tid; t < SITES; t += THREADS) {
    float* p1 = tF + 108 * NCOL + t * 18;
    float* p2 = wF + 12 * NCOL + t * 18;
    #pragma unroll
    for (int i = 0; i < 3; ++i)
      #pragma unroll
      for (int j = 0; j < 3; ++j) {
        float re = (i == j) ? 1.0f : 0.0f;
        p1[(i*3+j)*2] = re; p1[(i*3+j)*2 + 1] = 0.0f;
        p2[(i*3+j)*2] = re; p2[(i*3+j)*2 + 1] = 0.0f;
      }
  }

  __syncthreads();

  // ---- Phase B: rL[80x144] = Wt[80x112] x tF[112x144] via v_wmma_f32_16x16x4_f32
  {
    const int wv = tid >> 5, lane = tid & 31;
    const int l15 = lane & 15, kh = lane >> 4;   // A: M=l15; B/C/D: N=l15
    for (int tile = wv; tile < 5 * 9; tile += 8) {   // wave-uniform -> EXEC all 1s
      int mt = tile / 9, nt = tile % 9;
      v8f acc = {};
      const float* wbase = Wt + (mt * 16 + l15) * KPAD;
      const float* bbase = tF + nt * 16 + l15;
      for (int k0 = 0; k0 < KPAD; k0 += 4) {
        v2f a, b;
        // A 16x4 f32 layout: VGPR0 = K=(0|2), VGPR1 = K=(1|3) (lanes<16 | lanes>=16)
        a[0] = wbase[k0 + 2 * kh];
        a[1] = wbase[k0 + 2 * kh + 1];
        // B 4x16 f32 layout: row K striped across lanes per VGPR
        const float* bp = bbase + (k0 + 2 * kh) * NCOL;
        b[0] = bp[0];
        b[1] = bp[NCOL];
        acc = __builtin_amdgcn_wmma_f32_16x16x4_f32(
            false, a, false, b, (short)0, acc, false, false);
      }
      // C/D layout: VGPR q = M=q (lanes 0-15) / M=8+q (lanes 16-31), N = lane&15
      float* rp = rL + (mt * 16 + (lane >> 4) * 8) * NCOL + nt * 16 + l15;
      #pragma unroll
      for (int q = 0; q < 8; ++q) rp[q * NCOL] = acc[q];
    }
  }
  __syncthreads();

  // ---- Phase C: loops[u][i][k] = sum_{v,j} w_f[v][i][j] * r[u,v][j][k] ----
  for (int t = tid; t < SITES * 6 * 9; t += THREADS) {
    int ls = t / 54, rem = t % 54;
    int u = rem / 9, ik = rem % 9, i = ik / 3, k = ik % 3;
    float2 acc = make_float2(0.f, 0.f);
    #pragma unroll
    for (int v = 0; v < 13; ++v) {
      const float* wp = wF + v * NCOL + ls * 18;
      const float* rp = rL + (u * 13 + v) * NCOL + ls * 18;
      #pragma unroll
      for (int j = 0; j < 3; ++j) {
        float2 a = make_float2(wp[(i*3+j)*2], wp[(i*3+j)*2 + 1]);
        float2 b = make_float2(rp[(j*3+k)*2], rp[(j*3+k)*2 + 1]);
        acc.x += a.x * b.x - a.y * b.y;
        acc.y += a.x * b.y + a.y * b.x;
      }
    }
    int bx = base + ls;
    float* op = out + ((size_t)bx * 10 + 4 + u) * 18 + ik * 2;
    op[0] = acc.x; op[1] = acc.y;
  }

  // copy gauge-link channels 0..3 straight through
  for (int t = tid; t < SITES * 4 * 18; t += THREADS) {
    int ls = t / 72, rem = t % 72;
    int ch = rem / 18, e = rem % 18;
    size_t idx = ((size_t)(base + ls) * 10 + ch) * 18 + e;
    out[idx] = xin[idx];
  }
}

extern "C" void kernel_launch(void* const* d_in, const int* in_sizes, int n_in,
                              void* d_out, int out_size, void* d_ws, size_t ws_size,
                              hipStream_t stream) {
  (void)in_sizes; (void)n_in; (void)out_size; (void)d_ws; (void)ws_size;
  const float* x = (const float*)d_in[0];
  const float* w = (const float*)d_in[1];
  float* out = (float*)d_out;

  const size_t shmem =
      (size_t)(MPAD * KPAD + KPAD * NCOL + MPAD * NCOL + 13 * NCOL) * sizeof(float);
  (void)hipFuncSetAttribute((const void*)lconv_bilin_kernel,
                            hipFuncAttributeMaxDynamicSharedMemorySize, (int)shmem);
  lconv_bilin_kernel<<<NS_TOTAL / SITES, THREADS, shmem, stream>>>(x, w, out);
}